// UnsupervisedLoss_54408645706267
// MI455X (gfx1250) — compile-verified
//
#include <hip/hip_runtime.h>
#include <math.h>

typedef __attribute__((ext_vector_type(2))) float v2f;
typedef __attribute__((ext_vector_type(8))) float v8f;

#define BATCH 2
#define P 19248
#define K1 200
#define K2 100
#define H 256
#define W 256
#define GH 16
#define GW 16
#define PH 64
#define PW 64
#define EPS 1e-6f
#define VAR_XY 0.1f
#define VAR_WH 0.2f
#define NPAD 112            // 100 kept boxes padded to 7 WMMA n-tiles
#define NBLK_REC (BATCH * K2 * H)   // one block per (b, n, row) = 51200

// ---------------------------------------------------------------- helpers
__device__ __forceinline__ void decode_box(const float* lc, const float* pr,
                                           float& cx, float& cy, float& sx, float& sy) {
    cx = pr[0] + lc[0] * VAR_XY * pr[2];
    cy = pr[1] + lc[1] * VAR_XY * pr[3];
    float w = pr[2] * expf(lc[2] * VAR_WH);
    float h = pr[3] * expf(lc[3] * VAR_WH);
    sx = w * 0.5f + EPS;
    sy = h * 0.5f + EPS;
}

__device__ __forceinline__ float sigmoidf_(float x) { return 1.0f / (1.0f + expf(-x)); }

// ---------------------------------------------------------------- K1: fg prob
__global__ void k_softmax_fg(const float* __restrict__ conf, float* __restrict__ conffg) {
    int i = blockIdx.x * blockDim.x + threadIdx.x;
    if (i >= BATCH * P) return;
    float c0 = conf[2 * i + 0], c1 = conf[2 * i + 1];
    conffg[i] = 1.0f / (1.0f + expf(c0 - c1));  // softmax fg = sigmoid(c1-c0)
}

// ---------------------------------------------------------------- K2: top-200 (stable)
__global__ void k_top200(float* __restrict__ conffg, int* __restrict__ idx200,
                         float* __restrict__ sc200) {
    __shared__ float sv[256];
    __shared__ int   si[256];
    int b = blockIdx.x, t = threadIdx.x;
    float* cf = conffg + (size_t)b * P;
    for (int k = 0; k < K1; ++k) {
        float best = -1e30f; int bi = 0x7fffffff;
        for (int p = t; p < P; p += 256) {
            float v = cf[p];
            if (v > best || (v == best && p < bi)) { best = v; bi = p; }
        }
        sv[t] = best; si[t] = bi;
        __syncthreads();
        for (int s = 128; s > 0; s >>= 1) {
            if (t < s) {
                if (sv[t + s] > sv[t] || (sv[t + s] == sv[t] && si[t + s] < si[t])) {
                    sv[t] = sv[t + s]; si[t] = si[t + s];
                }
            }
            __syncthreads();
        }
        if (t == 0) {
            idx200[b * K1 + k] = si[0];
            sc200[b * K1 + k]  = sv[0];
            cf[si[0]] = -1e30f;
        }
        __syncthreads();
    }
}

// ---------------------------------------------------------------- K3: 16x16 gaussians
__global__ void k_gauss16(const float* __restrict__ loc, const float* __restrict__ priors,
                          const int* __restrict__ idx200, float* __restrict__ g16) {
    int blk = blockIdx.x;                 // b*K1 + i
    int b = blk / K1, i = blk % K1, t = threadIdx.x;
    int gi = idx200[b * K1 + i];
    float cx, cy, sx, sy;
    decode_box(loc + ((size_t)b * P + gi) * 4, priors + (size_t)gi * 4, cx, cy, sx, sy);
    int gx = t & (GW - 1), gy = t >> 4;
    float x = (gx + 0.5f) / GW, y = (gy + 0.5f) / GH;
    float dx = (x - cx) / sx, dy = (y - cy) / sy;
    g16[(size_t)blk * 256 + t] = expf(-0.5f * (dx * dx + dy * dy));
}

// ---------------------------------------------------------------- K4: iou_max[j] = max_{i<j} iou(i,j)
__global__ void k_ioumax(const float* __restrict__ g16, float* __restrict__ ioumax) {
    __shared__ float gj[256];
    __shared__ float red[256];
    int blk = blockIdx.x;                 // b*K1 + j
    int b = blk / K1, j = blk % K1, t = threadIdx.x;
    gj[t] = g16[(size_t)blk * 256 + t];
    __syncthreads();
    float iou = 0.0f;
    if (t < j) {                          // thread t handles pair (i=t, j)
        const float* gi = g16 + ((size_t)b * K1 + t) * 256;
        float smin = 0.0f, smax = 0.0f;
        for (int c = 0; c < 256; ++c) {
            float a = gi[c], bb = gj[c];
            smin += fminf(a, bb);
            smax += fmaxf(a, bb);
        }
        iou = smin / (smax + EPS);
    }
    red[t] = iou;
    __syncthreads();
    for (int s = 128; s > 0; s >>= 1) {
        if (t < s) red[t] = fmaxf(red[t], red[t + s]);
        __syncthreads();
    }
    if (t == 0) ioumax[b * K1 + j] = red[0];
}

// ---------------------------------------------------------------- K5: keep 100 + gather
__global__ void k_keep_gather(float* __restrict__ ioumax, const int* __restrict__ idx200,
                              const float* __restrict__ sc200,
                              const float* __restrict__ loc, const float* __restrict__ priors,
                              const float* __restrict__ mask,
                              int* __restrict__ keep, float* __restrict__ kmaskW,
                              float* __restrict__ params) {
    __shared__ float sv[256];
    __shared__ int   si[256];
    int b = blockIdx.x, t = threadIdx.x;
    float* im = ioumax + (size_t)b * K1;
    for (int n = 0; n < K2; ++n) {        // iterative argmin, tie -> lower index
        float best = 1e30f; int bi = 0x7fffffff;
        if (t < K1) { best = im[t]; bi = t; }
        sv[t] = best; si[t] = bi;
        __syncthreads();
        for (int s = 128; s > 0; s >>= 1) {
            if (t < s) {
                if (sv[t + s] < sv[t] || (sv[t + s] == sv[t] && si[t + s] < si[t])) {
                    sv[t] = sv[t + s]; si[t] = si[t + s];
                }
            }
            __syncthreads();
        }
        if (t == 0) { keep[b * K2 + n] = si[0]; im[si[0]] = 1e30f; }
        __syncthreads();
    }
    if (t < K2) {
        int kp = keep[b * K2 + t];
        int gi = idx200[b * K1 + kp];
        float cx, cy, sx, sy;
        decode_box(loc + ((size_t)b * P + gi) * 4, priors + (size_t)gi * 4, cx, cy, sx, sy);
        float* pr = params + ((size_t)b * K2 + t) * 8;
        pr[0] = cx; pr[1] = cy; pr[2] = sx; pr[3] = sy;
        pr[4] = sc200[b * K1 + kp];
        const float* ms = mask + ((size_t)b * P + gi) * 32;
        float* dst = kmaskW + ((size_t)b * NPAD + t) * 32;
        for (int m = 0; m < 32; ++m) dst[m] = ms[m];
    } else if (t < NPAD) {                // zero-pad rows 100..111 for WMMA tiles
        float* dst = kmaskW + ((size_t)b * NPAD + t) * 32;
        for (int m = 0; m < 32; ++m) dst[m] = 0.0f;
    }
}

// ---------------------------------------------------------------- K6: base = sigmoid(resize(proto_x) . W_dec)
__global__ void k_base(const float* __restrict__ proto_x, const float* __restrict__ Wd,
                       float* __restrict__ base) {
    int i = blockIdx.x * blockDim.x + threadIdx.x;
    if (i >= BATCH * H * W) return;
    int w = i % W, h = (i / W) % H, b = i / (H * W);
    // half-pixel bilinear 64 -> 256, clamp at edges
    float fy = (h + 0.5f) * 0.25f - 0.5f;
    float fx = (w + 0.5f) * 0.25f - 0.5f;
    float y0f = floorf(fy), x0f = floorf(fx);
    float wy = fy - y0f, wx = fx - x0f;
    int y0 = min(max((int)y0f, 0), PH - 1), y1 = min(y0 + ((int)y0f < 0 ? 0 : 1), PH - 1);
    int x0 = min(max((int)x0f, 0), PW - 1), x1 = min(x0 + ((int)x0f < 0 ? 0 : 1), PW - 1);
    if ((int)y0f < 0) wy = 0.0f;
    if ((int)x0f < 0) wx = 0.0f;
    const float* p00 = proto_x + (((size_t)b * PH + y0) * PW + x0) * 32;
    const float* p01 = proto_x + (((size_t)b * PH + y0) * PW + x1) * 32;
    const float* p10 = proto_x + (((size_t)b * PH + y1) * PW + x0) * 32;
    const float* p11 = proto_x + (((size_t)b * PH + y1) * PW + x1) * 32;
    float w00 = (1 - wy) * (1 - wx), w01 = (1 - wy) * wx, w10 = wy * (1 - wx), w11 = wy * wx;
    float acc0 = 0, acc1 = 0, acc2 = 0;
    for (int m = 0; m < 32; ++m) {
        float f = w00 * p00[m] + w01 * p01[m] + w10 * p10[m] + w11 * p11[m];
        acc0 += f * Wd[m * 3 + 0];
        acc1 += f * Wd[m * 3 + 1];
        acc2 += f * Wd[m * 3 + 2];
    }
    base[(((size_t)b * 3 + 0) * H + h) * W + w] = sigmoidf_(acc0);
    base[(((size_t)b * 3 + 1) * H + h) * W + w] = sigmoidf_(acc1);
    base[(((size_t)b * 3 + 2) * H + h) * W + w] = sigmoidf_(acc2);
}

// ---------------------------------------------------------------- K7: WMMA GEMM (proto . kmask^T) + gaussian epilogue
// A = proto[b, 16 pixels, 32]  (16x32 f32), B = kmaskW[b, 16 boxes, 32]^T (32x16),
// done as 8 chained V_WMMA_F32_16X16X4_F32 (K=4 each).
__global__ void k_wmma_mask(const float* __restrict__ proto, const float* __restrict__ kmaskW,
                            const float* __restrict__ params, float* __restrict__ mC64) {
    int b = blockIdx.z, mtile = blockIdx.x, ntile = blockIdx.y;
    int lane = threadIdx.x;
    int rc   = lane & 15;          // A row / B column handled by this lane
    int half = lane >> 4;          // K pair selector (lanes 16-31 hold K=2,3 of each step)
    const float* Ap = proto  + (((size_t)b * 4096 + mtile * 16 + rc) * 32) + 2 * half;
    const float* Bp = kmaskW + (((size_t)b * NPAD + ntile * 16 + rc) * 32) + 2 * half;
    v8f c = {};
    #pragma unroll
    for (int kk = 0; kk < 8; ++kk) {
        v2f a  = { Ap[4 * kk], Ap[4 * kk + 1] };
        v2f bf = { Bp[4 * kk], Bp[4 * kk + 1] };
        c = __builtin_amdgcn_wmma_f32_16x16x4_f32(false, a, false, bf, (short)0, c,
                                                  false, false);
    }
    // epilogue: D[row][col] with row = v + 8*half, col = rc  (ISA 16x16 f32 C/D layout)
    int n = ntile * 16 + rc;
    if (n < K2) {
        const float* pr = params + ((size_t)b * K2 + n) * 8;
        float cx = pr[0], cy = pr[1], sx = pr[2], sy = pr[3], cf = pr[4];
        #pragma unroll
        for (int v = 0; v < 8; ++v) {
            int row = v + 8 * half;
            int pix = mtile * 16 + row;
            int gx = pix & (PW - 1), gy = pix >> 6;
            float x = (gx + 0.5f) / PW, y = (gy + 0.5f) / PH;
            float dx = (x - cx) / sx, dy = (y - cy) / sy;
            float g = expf(-0.5f * (dx * dx + dy * dy));
            mC64[((size_t)b * K2 + n) * 4096 + pix] = sigmoidf_(c[v]) * g * cf;
        }
    }
}

// ---------------------------------------------------------------- K8: fused reconstruction loss
__global__ void k_recloss(const float* __restrict__ original, const float* __restrict__ base,
                          const float* __restrict__ mC64, const float* __restrict__ params,
                          float* __restrict__ partials) {
    __shared__ float red[256];
    int blk = blockIdx.x;                 // ((b*K2)+n)*H + h
    int h = blk % H, n = (blk / H) % K2, b = blk / (H * K2);
    int w = threadIdx.x;
    const float* pr = params + ((size_t)b * K2 + n) * 8;
    float cx = pr[0], cy = pr[1], sx = pr[2], sy = pr[3];
    // full-res gaussian
    float x = (w + 0.5f) / W, y = (h + 0.5f) / H;
    float dx = (x - cx) / sx, dy = (y - cy) / sy;
    float G = expf(-0.5f * (dx * dx + dy * dy));
    // bilinear upsample of mC64 (64 -> 256, half-pixel, clamp)
    float fy = (h + 0.5f) * 0.25f - 0.5f;
    float fx = (w + 0.5f) * 0.25f - 0.5f;
    float y0f = floorf(fy), x0f = floorf(fx);
    float wy = fy - y0f, wx = fx - x0f;
    int y0 = min(max((int)y0f, 0), PH - 1), y1 = min(y0 + ((int)y0f < 0 ? 0 : 1), PH - 1);
    int x0 = min(max((int)x0f, 0), PW - 1), x1 = min(x0 + ((int)x0f < 0 ? 0 : 1), PW - 1);
    if ((int)y0f < 0) wy = 0.0f;
    if ((int)x0f < 0) wx = 0.0f;
    const float* mrow = mC64 + ((size_t)b * K2 + n) * 4096;
    float mC = (1 - wy) * ((1 - wx) * mrow[y0 * PW + x0] + wx * mrow[y0 * PW + x1]) +
               wy       * ((1 - wx) * mrow[y1 * PW + x0] + wx * mrow[y1 * PW + x1]);
    float acc = 0.0f;
    #pragma unroll
    for (int cch = 0; cch < 3; ++cch) {
        size_t off = (((size_t)b * 3 + cch) * H + h) * W + w;
        float r = base[off] * G - original[off];
        acc += r * r;
    }
    red[threadIdx.x] = acc * mC;
    __syncthreads();
    for (int s = 128; s > 0; s >>= 1) {
        if (threadIdx.x < s) red[threadIdx.x] += red[threadIdx.x + s];
        __syncthreads();
    }
    if (threadIdx.x == 0) partials[blk] = red[0];
}

// ---------------------------------------------------------------- K9: finalize
__global__ void k_finalize(const float* __restrict__ partials, float* __restrict__ out) {
    __shared__ float red[256];
    int t = threadIdx.x;
    float acc = 0.0f;
    for (int i = t; i < NBLK_REC; i += 256) acc += partials[i];
    red[t] = acc;
    __syncthreads();
    for (int s = 128; s > 0; s >>= 1) {
        if (t < s) red[t] += red[t + s];
        __syncthreads();
    }
    if (t == 0) {
        out[0] = 0.0f;                                    // backgroundLoss == 0 identically
        out[1] = red[0] / (float)((size_t)BATCH * K2 * H * W);
    }
}

// ---------------------------------------------------------------- host
extern "C" void kernel_launch(void* const* d_in, const int* in_sizes, int n_in,
                              void* d_out, int out_size, void* d_ws, size_t ws_size,
                              hipStream_t stream) {
    const float* original = (const float*)d_in[0];
    const float* conf     = (const float*)d_in[1];
    const float* loc      = (const float*)d_in[2];
    const float* mask     = (const float*)d_in[3];
    const float* priors   = (const float*)d_in[4];
    const float* proto    = (const float*)d_in[5];
    const float* proto_x  = (const float*)d_in[6];
    const float* W_dec    = (const float*)d_in[7];
    float* out = (float*)d_out;
    float* ws  = (float*)d_ws;

    size_t off = 0;
    auto alloc = [&](size_t n) { size_t o = off; off += (n + 15) & ~(size_t)15; return o; };
    size_t o_conffg  = alloc((size_t)BATCH * P);
    size_t o_idx200  = alloc((size_t)BATCH * K1);          // int
    size_t o_sc200   = alloc((size_t)BATCH * K1);
    size_t o_g16     = alloc((size_t)BATCH * K1 * 256);
    size_t o_ioumax  = alloc((size_t)BATCH * K1);
    size_t o_keep    = alloc((size_t)BATCH * K2);          // int
    size_t o_kmask   = alloc((size_t)BATCH * NPAD * 32);
    size_t o_params  = alloc((size_t)BATCH * K2 * 8);
    size_t o_base    = alloc((size_t)BATCH * 3 * H * W);
    size_t o_mC64    = alloc((size_t)BATCH * K2 * 4096);
    size_t o_part    = alloc((size_t)NBLK_REC);
    (void)ws_size; (void)in_sizes; (void)n_in; (void)out_size;

    float* conffg = ws + o_conffg;
    int*   idx200 = (int*)(ws + o_idx200);
    float* sc200  = ws + o_sc200;
    float* g16    = ws + o_g16;
    float* ioumax = ws + o_ioumax;
    int*   keep   = (int*)(ws + o_keep);
    float* kmaskW = ws + o_kmask;
    float* params = ws + o_params;
    float* base   = ws + o_base;
    float* mC64   = ws + o_mC64;
    float* part   = ws + o_part;

    k_softmax_fg<<<(BATCH * P + 255) / 256, 256, 0, stream>>>(conf, conffg);
    k_top200<<<BATCH, 256, 0, stream>>>(conffg, idx200, sc200);
    k_gauss16<<<BATCH * K1, 256, 0, stream>>>(loc, priors, idx200, g16);
    k_ioumax<<<BATCH * K1, 256, 0, stream>>>(g16, ioumax);
    k_keep_gather<<<BATCH, 256, 0, stream>>>(ioumax, idx200, sc200, loc, priors, mask,
                                             keep, kmaskW, params);
    k_base<<<(BATCH * H * W + 255) / 256, 256, 0, stream>>>(proto_x, W_dec, base);
    dim3 wgrid(4096 / 16, NPAD / 16, BATCH);               // 256 x 7 x 2 waves
    k_wmma_mask<<<wgrid, 32, 0, stream>>>(proto, kmaskW, params, mC64);
    k_recloss<<<NBLK_REC, 256, 0, stream>>>(original, base, mC64, params, part);
    k_finalize<<<1, 256, 0, stream>>>(part, out);
}